// SSIMLoss_28698971472176
// MI455X (gfx1250) — compile-verified
//
#include <hip/hip_runtime.h>

// ---------------------------------------------------------------------------
// SSIM loss for [16,3,512,512] f32 via separable 11-tap Gaussian blur mapped
// onto V_WMMA_F32_16X16X4_F32 band-matrix matmuls. Memory-bound (~100 MB in,
// 23.3 TB/s -> ~4.3us floor), so f32 end-to-end is the right precision.
// ---------------------------------------------------------------------------

#define HALO     5
#define TILE     16
#define SROWS    26            // TILE + 2*HALO
#define SCOLS    28            // K-dim padded 26 -> 28 (multiple of 4)
#define TROWS    28
#define IMG      512
#define PLANES   48            // 16 batch * 3 channels
#define TPS      32            // tiles per side: 512/16
#define NTILES   (PLANES * TPS * TPS)   // 49152
#define WAVES_PB 8

typedef float v2f __attribute__((ext_vector_type(2)));
typedef float v8f __attribute__((ext_vector_type(8)));

// Normalized Gaussian taps, sigma = 1.5, window 11.
__constant__ float GW[11] = {
    0.00102838f, 0.00759874f, 0.03600078f, 0.10936072f, 0.21300555f,
    0.26601172f,
    0.21300555f, 0.10936072f, 0.03600078f, 0.00759874f, 0.00102838f
};

struct WaveLds {
    float p[SROWS * SCOLS];    // pred halo tile
    float t[SROWS * SCOLS];    // target halo tile
    float T[TROWS * TILE];     // horizontally blurred intermediate
};

__global__ __launch_bounds__(256) void ssim_wmma_kernel(
    const float* __restrict__ pred, const float* __restrict__ target,
    double* __restrict__ acc_out)
{
    __shared__ WaveLds lds[WAVES_PB];
    __shared__ float   bsum[WAVES_PB];

    const int lane  = threadIdx.x & 31;
    const int wave  = threadIdx.x >> 5;
    const int tile  = blockIdx.x * WAVES_PB + wave;

    const int plane = tile >> 10;            // / (32*32)
    const int tin   = tile & 1023;
    const int y0    = (tin >> 5) * TILE;
    const int x0    = (tin & 31) * TILE;

    const float* __restrict__ pb = pred   + (size_t)plane * (IMG * IMG);
    const float* __restrict__ tb = target + (size_t)plane * (IMG * IMG);
    WaveLds& L = lds[wave];

    // ---- stage halo tiles (zero-fill OOB: matches conv's zero padding) ----
    for (int idx = lane; idx < SROWS * SCOLS; idx += 32) {
        const int r  = idx / SCOLS;
        const int c  = idx - r * SCOLS;
        const int gy = y0 - HALO + r;
        const int gx = x0 - HALO + c;
        float pv = 0.f, tv = 0.f;
        if (gy >= 0 && gy < IMG && gx >= 0 && gx < IMG) {
            const int o = gy * IMG + gx;
            pv = pb[o];
            tv = tb[o];
        }
        L.p[idx] = pv;
        L.t[idx] = tv;
    }

    // ---- band-matrix registers: band(k - n) = GW[k-n] for 0<=k-n<=10 ----
    // Per ISA layout these lane values serve BOTH as B (pass 1, 4x16 blocks)
    // and as A (pass 2, 16x4 blocks): k = 4*kk + 2*(lane>>4) + comp, n = lane&15.
    const int half2 = (lane >> 4) << 1;      // 0 or 2
    const int j     = lane & 15;
    v2f W[7];
#pragma unroll
    for (int kk = 0; kk < 7; ++kk) {
        const int k0 = kk * 4 + half2;
        const int d0 = k0 - j, d1 = k0 + 1 - j;
        W[kk].x = (d0 >= 0 && d0 <= 10) ? GW[d0] : 0.f;
        W[kk].y = (d1 >= 0 && d1 <= 10) ? GW[d1] : 0.f;
    }

    // zero T pad rows 26..27 once (band weight there is 0; keep finite)
    L.T[(26 + (lane >> 4)) * TILE + j] = 0.f;

    v8f res[5];
#pragma unroll
    for (int s = 0; s < 5; ++s) {
        // ---- pass 1: horizontal blur -> T rows 0..25 (two 16-row tiles) ----
#pragma unroll
        for (int dt = 0; dt < 2; ++dt) {
            const int r0 = dt * 10;
            const int m  = r0 + j;           // A row (S tile row)
            v8f acc = {};
#pragma unroll
            for (int kk = 0; kk < 7; ++kk) {
                const int o = m * SCOLS + kk * 4 + half2;
                const float p0 = L.p[o], p1 = L.p[o + 1];
                const float t0 = L.t[o], t1 = L.t[o + 1];
                v2f a;
                switch (s) {
                    case 0:  a.x = p0;      a.y = p1;      break;
                    case 1:  a.x = t0;      a.y = t1;      break;
                    case 2:  a.x = p0 * p0; a.y = p1 * p1; break;
                    case 3:  a.x = t0 * t0; a.y = t1 * t1; break;
                    default: a.x = p0 * t0; a.y = p1 * t1; break;
                }
                acc = __builtin_amdgcn_wmma_f32_16x16x4_f32(
                    false, a, false, W[kk], (short)0, acc, false, false);
            }
            const int tr = r0 + (lane >> 4) * 8;   // D row base for this lane
#pragma unroll
            for (int v = 0; v < 8; ++v)
                L.T[(tr + v) * TILE + j] = acc[v];
        }

        // cross-lane LDS RAW inside the wave: drain DS counter before reads
        asm volatile("s_wait_dscnt 0x0" ::: "memory");

        // ---- pass 2: vertical blur (band as A, T as B) ----
        {
            v8f acc = {};
#pragma unroll
            for (int kk = 0; kk < 7; ++kk) {
                const int k0 = kk * 4 + half2;
                v2f b;
                b.x = L.T[k0 * TILE + j];
                b.y = L.T[(k0 + 1) * TILE + j];
                acc = __builtin_amdgcn_wmma_f32_16x16x4_f32(
                    false, W[kk], false, b, (short)0, acc, false, false);
            }
            res[s] = acc;
        }
    }

    // ---- per-pixel SSIM on the five co-laid-out accumulators ----
    const float C1c = 0.0001f, C2c = 0.0009f;
    float sum = 0.f;
#pragma unroll
    for (int e = 0; e < 8; ++e) {
        const float mp  = res[0][e], mt  = res[1][e];
        const float ep2 = res[2][e], et2 = res[3][e], ept = res[4][e];
        const float mp2 = mp * mp, mt2 = mt * mt, mpt = mp * mt;
        const float sp  = ep2 - mp2;
        const float st  = et2 - mt2;
        const float spt = ept - mpt;
        const float num = (2.f * mpt + C1c) * (2.f * spt + C2c);
        const float den = (mp2 + mt2 + C1c) * (sp + st + C2c);
        sum += num / den;
    }

    // ---- wave32 reduce, then block reduce, then one f64 atomic per block ----
#pragma unroll
    for (int off = 16; off > 0; off >>= 1)
        sum += __shfl_xor(sum, off, 32);
    if (lane == 0) bsum[wave] = sum;
    __syncthreads();
    if (threadIdx.x == 0) {
        double tot = 0.0;
#pragma unroll
        for (int w = 0; w < WAVES_PB; ++w) tot += (double)bsum[w];
        atomicAdd(acc_out, tot);
    }
}

__global__ void ssim_zero_kernel(double* a) { a[0] = 0.0; }

__global__ void ssim_finalize_kernel(const double* a, float* out) {
    out[0] = 1.0f - (float)(a[0] / (double)(16.0 * 3.0 * 512.0 * 512.0));
}

extern "C" void kernel_launch(void* const* d_in, const int* in_sizes, int n_in,
                              void* d_out, int out_size, void* d_ws, size_t ws_size,
                              hipStream_t stream) {
    const float* pred   = (const float*)d_in[0];
    const float* target = (const float*)d_in[1];
    double* acc = (double*)d_ws;

    ssim_zero_kernel<<<1, 1, 0, stream>>>(acc);
    ssim_wmma_kernel<<<NTILES / WAVES_PB, 256, 0, stream>>>(pred, target, acc);
    ssim_finalize_kernel<<<1, 1, 0, stream>>>(acc, (float*)d_out);
}